// GNNStructEncoder_83906481095127
// MI455X (gfx1250) — compile-verified
//
#include <hip/hip_runtime.h>

// ---------------------------------------------------------------------------
// GIN 2-layer GNN for MI455X (gfx1250, wave32).
// GEMMs run on the WMMA pipe with V_WMMA_F32_16X16X4_F32 (full f32 precision;
// workload is HBM-bound at ~2.4 FLOP/B so low-precision WMMA buys nothing).
// ---------------------------------------------------------------------------

typedef float v2f __attribute__((ext_vector_type(2)));
typedef float v8f __attribute__((ext_vector_type(8)));

#define GIN_D 128

// ---------------------------------------------------------------- zero fill
__global__ void __launch_bounds__(256) k_zero(float* __restrict__ p, long n) {
  long i = (long)blockIdx.x * blockDim.x + threadIdx.x;
  long stride = (long)gridDim.x * blockDim.x;
  for (; i < n; i += stride) p[i] = 0.0f;
}

// ------------------------------------------------- edge scatter-add (agg)
// One wave per edge: lane c handles float4 chunk c of the 128-float row.
// Coalesced 512B gather of h[src], native global_atomic_add_f32 into agg[dst].
__global__ void __launch_bounds__(256) k_scatter_add(
    const float* __restrict__ x, const int* __restrict__ src,
    const int* __restrict__ dst, float* __restrict__ agg, long nedges) {
  long total = nedges * 32;  // 32 float4 chunks per row
  long i = (long)blockIdx.x * blockDim.x + threadIdx.x;
  long stride = (long)gridDim.x * blockDim.x;
  for (; i < total; i += stride) {
    long e = i >> 5;
    int c = (int)(i & 31) << 2;
    int s = src[e];
    int d = dst[e];
    float4 v = *(const float4*)(x + (long)s * GIN_D + c);
    float* o = agg + (long)d * GIN_D + c;
    unsafeAtomicAdd(o + 0, v.x);
    unsafeAtomicAdd(o + 1, v.y);
    unsafeAtomicAdd(o + 2, v.z);
    unsafeAtomicAdd(o + 3, v.w);
  }
}

// ------------------------------------------------------------- WMMA GEMM
// C[nrows x 128] = act( (A (+ A2)) @ W + bias ),  W is 128x128 row-major.
// One wave computes a 16-row x 128-col tile:
//   8 column tiles x (K=128 / 4) v_wmma_f32_16x16x4_f32 steps.
// W^T staged in LDS (exactly 64KB) with XOR swizzle k ^ (4*(n&31)):
//   lanes 0-15 hit banks == {kb,kb+1} mod 4 group, lanes 16-31 the other
//   group -> conflict-free ds_load_b64 B-fragment reads.
// A-fragment layout (16x4 f32 A): lane l -> M = l&15, K = k0 + 2*(l>>4)+{0,1}.
// C layout: vgpr r, lanes 0-15 -> M = r, lanes 16-31 -> M = r+8; N = l&15.
__global__ void __launch_bounds__(256) k_gin_gemm(
    const float* __restrict__ A, const float* __restrict__ A2,
    const float* __restrict__ W, const float* __restrict__ bias,
    float* __restrict__ C, int nrows, int fuse_add, int do_relu) {
  __shared__ float sW[GIN_D * GIN_D];  // 64 KB, swizzled W^T

  int tid = threadIdx.x;
  for (int idx = tid; idx < GIN_D * GIN_D; idx += 256) {
    int k = idx >> 7;
    int n = idx & 127;
    sW[n * GIN_D + (k ^ ((n & 31) << 2))] = W[idx];
  }
  __syncthreads();

  int wave = tid >> 5;
  int lane = tid & 31;
  int tile = blockIdx.x * 8 + wave;          // wave-uniform
  if (tile * 16 >= nrows) return;            // whole-wave exit: EXEC stays ~0

  int m0 = tile << 4;
  int lm = lane & 15;               // M (for A) / N (for B,C) within tile
  int lk = (lane >> 4) << 1;        // K sub-offset: 0 or 2

  // Preload the full 16x128 A tile as fragments (fused z = A + A2).
  const float* arow = A + (size_t)(m0 + lm) * GIN_D;
  v2f afrag[32];
  if (fuse_add) {
    const float* arow2 = A2 + (size_t)(m0 + lm) * GIN_D;
#pragma unroll
    for (int ks = 0; ks < 32; ++ks) {
      int k = (ks << 2) + lk;
      v2f a = *(const v2f*)(arow + k);
      v2f b = *(const v2f*)(arow2 + k);
      a.x += b.x;
      a.y += b.y;
      afrag[ks] = a;
    }
  } else {
#pragma unroll
    for (int ks = 0; ks < 32; ++ks) {
      int k = (ks << 2) + lk;
      afrag[ks] = *(const v2f*)(arow + k);
    }
  }

  v8f acc[8];
#pragma unroll
  for (int t = 0; t < 8; ++t)
#pragma unroll
    for (int r = 0; r < 8; ++r) acc[t][r] = 0.0f;

#pragma unroll
  for (int ks = 0; ks < 32; ++ks) {
    int kb = (ks << 2) + lk;
#pragma unroll
    for (int t = 0; t < 8; ++t) {
      int n = t * 16 + lm;
      v2f b = *(const v2f*)(&sW[n * GIN_D + (kb ^ ((n & 31) << 2))]);
      acc[t] = __builtin_amdgcn_wmma_f32_16x16x4_f32(
          false, afrag[ks], false, b, (short)0, acc[t], false, false);
    }
  }

  // Epilogue: bias (+ReLU), store per C layout.
  int rbase = m0 + ((lane >> 4) << 3);
#pragma unroll
  for (int t = 0; t < 8; ++t) {
    int n = t * 16 + lm;
    float bv = bias[n];
#pragma unroll
    for (int r = 0; r < 8; ++r) {
      float v = acc[t][r] + bv;
      if (do_relu) v = fmaxf(v, 0.0f);
      C[(size_t)(rbase + r) * GIN_D + n] = v;
    }
  }
}

// ----------------------------------------------- PairNorm statistics pass
// stats[0..127] += column sums, stats[128] += total sum of squares.
__global__ void __launch_bounds__(256) k_pn_stats(
    const float* __restrict__ x, float* __restrict__ stats, int n) {
  __shared__ float s[256];
  int col = threadIdx.x & 127;
  int half = threadIdx.x >> 7;  // two rows in flight per block
  float csum = 0.0f, sq = 0.0f;
  for (long row = (long)blockIdx.x * 2 + half; row < n;
       row += (long)gridDim.x * 2) {
    float v = x[row * GIN_D + col];
    csum += v;
    sq += v * v;
  }
  s[threadIdx.x] = csum;
  __syncthreads();
  if (half == 0) unsafeAtomicAdd(&stats[col], csum + s[col + 128]);
  __syncthreads();
  s[threadIdx.x] = sq;
  __syncthreads();
  for (int off = 128; off > 0; off >>= 1) {
    if (threadIdx.x < off) s[threadIdx.x] += s[threadIdx.x + off];
    __syncthreads();
  }
  if (threadIdx.x == 0) unsafeAtomicAdd(&stats[GIN_D], s[0]);
}

// mu_inv[j] = column mean, mu_inv[128] = 1 / sqrt(eps + mean row-norm^2)
// using  sum_ij (x-mu)^2 = Q - sum_j S_j^2 / N.
__global__ void __launch_bounds__(128) k_pn_finalize(
    const float* __restrict__ stats, float* __restrict__ mu_inv, int n) {
  __shared__ float s[128];
  int j = threadIdx.x;
  float invn = 1.0f / (float)n;
  float S = stats[j];
  float mu = S * invn;
  mu_inv[j] = mu;
  s[j] = mu * S;  // S_j^2 / N
  __syncthreads();
  for (int off = 64; off > 0; off >>= 1) {
    if (j < off) s[j] += s[j + off];
    __syncthreads();
  }
  if (j == 0) {
    float ss = stats[GIN_D] - s[0];
    mu_inv[GIN_D] = 1.0f / sqrtf(1e-6f + ss * invn);
  }
}

// y = relu((x - mu[col]) * inv)
__global__ void __launch_bounds__(256) k_pn_norm(
    const float* __restrict__ x, const float* __restrict__ mu_inv,
    float* __restrict__ y, long n) {
  float inv = mu_inv[GIN_D];
  long i = (long)blockIdx.x * blockDim.x + threadIdx.x;
  long stride = (long)gridDim.x * blockDim.x;
  for (; i < n; i += stride) {
    int col = (int)(i & 127);
    float v = (x[i] - mu_inv[col]) * inv;
    y[i] = fmaxf(v, 0.0f);
  }
}

// ---------------------------------------------------------------------------
extern "C" void kernel_launch(void* const* d_in, const int* in_sizes, int n_in,
                              void* d_out, int out_size, void* d_ws,
                              size_t ws_size, hipStream_t stream) {
  const float* h = (const float*)d_in[0];
  const int* src = (const int*)d_in[1];
  const int* dst = (const int*)d_in[2];
  const float* W1a = (const float*)d_in[3];
  const float* b1a = (const float*)d_in[4];
  const float* W1b = (const float*)d_in[5];
  const float* b1b = (const float*)d_in[6];
  const float* W4a = (const float*)d_in[7];
  const float* b4a = (const float*)d_in[8];
  const float* W4b = (const float*)d_in[9];
  const float* b4b = (const float*)d_in[10];

  const int N = in_sizes[0] / GIN_D;
  const long E = in_sizes[1];
  const long ND = (long)N * GIN_D;

  // workspace: bufA (agg / l1), bufB (hidden), small stats buffers
  float* bufA = (float*)d_ws;
  float* bufB = bufA + ND;
  float* stats = bufB + ND;        // 129 floats
  float* mu_inv = stats + 160;     // 129 floats

  float* out_l5 = (float*)d_out;       // first output
  float* out_l1n = out_l5 + ND;        // second output (l1_norm)

  int tiles = (N + 15) / 16;
  int gemm_blocks = (tiles + 7) / 8;   // 8 waves (16-row tiles) per block

  // ---- conv1: agg = scatter(h); hidden = relu((h+agg)W1a+b1a); l1 = hidden*W1b+b1b
  k_zero<<<2048, 256, 0, stream>>>(bufA, ND);
  k_scatter_add<<<2048, 256, 0, stream>>>(h, src, dst, bufA, E);
  k_gin_gemm<<<gemm_blocks, 256, 0, stream>>>(h, bufA, W1a, b1a, bufB, N, 1, 1);
  k_gin_gemm<<<gemm_blocks, 256, 0, stream>>>(bufB, nullptr, W1b, b1b, bufA, N,
                                              0, 0);  // l1 -> bufA

  // ---- pairnorm + relu -> out_l1n
  k_zero<<<1, 256, 0, stream>>>(stats, 129);
  k_pn_stats<<<1024, 256, 0, stream>>>(bufA, stats, N);
  k_pn_finalize<<<1, 128, 0, stream>>>(stats, mu_inv, N);
  k_pn_norm<<<2048, 256, 0, stream>>>(bufA, mu_inv, out_l1n, ND);

  // ---- conv2 on l1_norm -> out_l5
  k_zero<<<2048, 256, 0, stream>>>(bufA, ND);
  k_scatter_add<<<2048, 256, 0, stream>>>(out_l1n, src, dst, bufA, E);
  k_gin_gemm<<<gemm_blocks, 256, 0, stream>>>(out_l1n, bufA, W4a, b4a, bufB, N,
                                              1, 1);
  k_gin_gemm<<<gemm_blocks, 256, 0, stream>>>(bufB, nullptr, W4b, b4b, out_l5,
                                              N, 0, 0);
}